// LightUnit_trans_40827959116378
// MI455X (gfx1250) — compile-verified
//
#include <hip/hip_runtime.h>
#include <hip/hip_bf16.h>

typedef __attribute__((ext_vector_type(2))) float v2f;
typedef __attribute__((ext_vector_type(8))) float v8f;

#define LU_EPS 1e-5f
// B=4 C=64 D=16 H=64 W=64, C8=8, k = int(0.6*64) = 38
// block: 8x16 spatial tile, all 64 channels; 128 threads (4 wave32)

__device__ __forceinline__ v8f wmma_f32_4(v2f a, v2f b, v8f c) {
  return __builtin_amdgcn_wmma_f32_16x16x4_f32(false, a, false, b, (short)0, c, false, false);
}

// SiLU with v_rcp_f32 (avoids IEEE div_scale/div_fmas/div_fixup chains)
__device__ __forceinline__ float silu_f(float z) {
  return z * __builtin_amdgcn_rcpf(1.0f + __expf(-z));
}

__global__ __launch_bounds__(128)
void lightunit_fused_kernel(const float* __restrict__ x,
                            const float* __restrict__ w_dw,
                            const float* __restrict__ bn1_gamma, const float* __restrict__ bn1_beta,
                            const float* __restrict__ bn1_mean,  const float* __restrict__ bn1_var,
                            const float* __restrict__ w_pw1, const float* __restrict__ b_pw1,
                            const float* __restrict__ bn2_gamma, const float* __restrict__ bn2_beta,
                            const float* __restrict__ bn2_mean,  const float* __restrict__ bn2_var,
                            const float* __restrict__ w_pw2, const float* __restrict__ b_pw2,
                            const float* __restrict__ alpha_p,
                            float* __restrict__ out)
{
  // LDS: 3840 + 8192 + 1024 + 1600 + 64 + 512 + 8 + 512 + 64 = 15816 floats = 63264 B
  __shared__ float xs[16 * 12 * 20];    // one channel-group halo tile
  __shared__ float attn_s[64 * 128];    // y1 then attn, channel-major
  __shared__ float hid_s[8 * 128];      // silu'd hidden
  __shared__ float wdws[64 * 25];       // bn1-folded depthwise weights
  __shared__ float shift1_s[64];
  __shared__ float w1f[8 * 64];         // bn2-folded pw1
  __shared__ float bias2f_s[8];
  __shared__ float w2s[64 * 8];
  __shared__ float b2s[64];

  const int t    = threadIdx.x;
  const int lane = t & 31;
  const int wv   = t >> 5;

  const int bid   = blockIdx.x;
  const int tile  = bid & 31;     // 32 tiles per (b,d) slice
  const int slice = bid >> 5;     // b*16 + d
  const int b     = slice >> 4;
  const int d     = slice & 15;
  const int h0    = (tile >> 2) * 8;
  const int w0    = (tile & 3) * 16;

  // ---------- phase 0: fold BN into weights, stage to LDS ----------
  for (int i = t; i < 1600; i += 128) {
    int c = i / 25;
    float s1 = bn1_gamma[c] * rsqrtf(bn1_var[c] + LU_EPS);
    wdws[i] = w_dw[i] * s1;
  }
  if (t < 64) {
    float s1 = bn1_gamma[t] * rsqrtf(bn1_var[t] + LU_EPS);
    shift1_s[t] = bn1_beta[t] - bn1_mean[t] * s1;
    b2s[t] = b_pw2[t];
  }
  for (int i = t; i < 512; i += 128) {
    int o = i >> 6;
    float s2 = bn2_gamma[o] * rsqrtf(bn2_var[o] + LU_EPS);
    w1f[i] = w_pw1[i] * s2;
    w2s[i] = w_pw2[i];
  }
  if (t < 8) {
    float s2 = bn2_gamma[t] * rsqrtf(bn2_var[t] + LU_EPS);
    bias2f_s[t] = (b_pw1[t] - bn2_mean[t]) * s2 + bn2_beta[t];
  }
  __syncthreads();

  const int m_a = lane & 15;   // WMMA A row
  const int hi  = lane >> 4;   // upper-half-of-wave flag
  const int n_b = lane & 15;   // WMMA B/D column

  v8f acch0 = {};  // pw1 accumulators: two 16-location N-tiles per wave
  v8f acch1 = {};

  // ---------- phase 1: channel-group loop (4 x 16 channels) ----------
  for (int g = 0; g < 4; ++g) {
    // (a) cooperative load of 16-channel halo tile (zero-padded)
    const float* xg = x + (((size_t)(b * 64 + g * 16) * 16 + d) * 4096);
    for (int i = t; i < 16 * 12 * 20; i += 128) {
      int cl = i / 240;
      int rem = i % 240;
      int r = rem / 20;
      int col = rem % 20;
      int gh = h0 - 2 + r;
      int gw = w0 - 2 + col;
      float val = 0.0f;
      if (gh >= 0 && gh < 64 && gw >= 0 && gw < 64)
        val = xg[(size_t)cl * (16 * 4096) + gh * 64 + gw];
      xs[i] = val;
    }
    __syncthreads();

    // prefetch next channel-group's halo rows (global_prefetch) while we compute
    if (g < 3) {
      const float* xn = x + (((size_t)(b * 64 + (g + 1) * 16) * 16 + d) * 4096);
      for (int i = t; i < 16 * 12; i += 128) {
        int cl = i / 12;
        int r  = i % 12;
        int gh = h0 - 2 + r;
        if (gh >= 0 && gh < 64) {
          int gw = (w0 - 2 < 0) ? 0 : (w0 - 2);
          const float* p = &xn[(size_t)cl * (16 * 4096) + gh * 64 + gw];
          __builtin_prefetch(p, 0, 3);
          __builtin_prefetch(p + 16, 0, 3);
        }
      }
    }

    // (b) depthwise 5x5 conv, one (channel,row) strip per thread
    {
      int cl  = t >> 3;        // 0..15 local channel
      int row = t & 7;         // 0..7 output row
      int c   = g * 16 + cl;
      float wt[25];
      #pragma unroll
      for (int i2 = 0; i2 < 25; ++i2) wt[i2] = wdws[c * 25 + i2];
      float acc[16];
      #pragma unroll
      for (int w = 0; w < 16; ++w) acc[w] = 0.0f;
      #pragma unroll
      for (int kr = 0; kr < 5; ++kr) {
        float rb[20];
        #pragma unroll
        for (int j = 0; j < 20; ++j) rb[j] = xs[cl * 240 + (row + kr) * 20 + j];
        #pragma unroll
        for (int w = 0; w < 16; ++w) {
          #pragma unroll
          for (int kc = 0; kc < 5; ++kc)
            acc[w] = fmaf(rb[w + kc], wt[kr * 5 + kc], acc[w]);
        }
      }
      float sh = shift1_s[c];
      #pragma unroll
      for (int w = 0; w < 16; ++w)
        attn_s[c * 128 + row * 16 + w] = silu_f(acc[w] + sh);
    }

    // (c) pw1 partial GEMM: hidden += W1f[:, g16:g16+16] @ x_g  (f32 WMMA)
    float amask = (m_a < 8) ? 1.0f : 0.0f;   // pad M 8->16 with zero rows
    int arow = (m_a & 7) * 64;
    #pragma unroll
    for (int kk = 0; kk < 4; ++kk) {
      int kbase = g * 16 + kk * 4;
      v2f a;
      a[0] = w1f[arow + kbase + 2 * hi + 0] * amask;
      a[1] = w1f[arow + kbase + 2 * hi + 1] * amask;
      {
        int nt = wv * 2 + 0;
        v2f bb;
        bb[0] = xs[(kk * 4 + 0 + 2 * hi) * 240 + (nt + 2) * 20 + (n_b + 2)];
        bb[1] = xs[(kk * 4 + 1 + 2 * hi) * 240 + (nt + 2) * 20 + (n_b + 2)];
        acch0 = wmma_f32_4(a, bb, acch0);
      }
      {
        int nt = wv * 2 + 1;
        v2f bb;
        bb[0] = xs[(kk * 4 + 0 + 2 * hi) * 240 + (nt + 2) * 20 + (n_b + 2)];
        bb[1] = xs[(kk * 4 + 1 + 2 * hi) * 240 + (nt + 2) * 20 + (n_b + 2)];
        acch1 = wmma_f32_4(a, bb, acch1);
      }
    }
    __syncthreads();  // before next group's xs overwrite
  }

  // ---------- phase 2: bias + SiLU on hidden, write to LDS ----------
  if (lane < 16) {
    int n = lane;
    #pragma unroll
    for (int s = 0; s < 2; ++s) {
      int loc = (wv * 2 + s) * 16 + n;
      v8f av = s ? acch1 : acch0;
      #pragma unroll
      for (int r = 0; r < 8; ++r)
        hid_s[r * 128 + loc] = silu_f(av[r] + bias2f_s[r]);
    }
  }
  __syncthreads();

  // ---------- phase 3: pw2 GEMM (64x128 = W2[64x8] @ hid[8x128]), attn = y1*y2 ----------
  for (int tile2 = wv; tile2 < 32; tile2 += 4) {
    int mt = tile2 >> 3;
    int nt = tile2 & 7;
    v8f acc = {};
    #pragma unroll
    for (int kk = 0; kk < 2; ++kk) {
      v2f a;
      a[0] = w2s[(mt * 16 + m_a) * 8 + kk * 4 + 2 * hi + 0];
      a[1] = w2s[(mt * 16 + m_a) * 8 + kk * 4 + 2 * hi + 1];
      v2f bb;
      bb[0] = hid_s[(kk * 4 + 0 + 2 * hi) * 128 + nt * 16 + n_b];
      bb[1] = hid_s[(kk * 4 + 1 + 2 * hi) * 128 + nt * 16 + n_b];
      acc = wmma_f32_4(a, bb, acc);
    }
    #pragma unroll
    for (int r = 0; r < 8; ++r) {
      int c = mt * 16 + r + 8 * hi;
      int loc = nt * 16 + n_b;
      float y2v = acc[r] + b2s[c];
      attn_s[c * 128 + loc] *= y2v;   // y1 * y2
    }
  }
  __syncthreads();

  // ---------- phase 4: per-voxel top-k(38) threshold + masked residual ----------
  {
    float v[64];
    #pragma unroll
    for (int c = 0; c < 64; ++c) v[c] = attn_s[c * 128 + t];
    // fully-unrolled 64-element bitonic sort (ascending)
    #pragma unroll
    for (int kk = 2; kk <= 64; kk <<= 1) {
      #pragma unroll
      for (int j = kk >> 1; j > 0; j >>= 1) {
        #pragma unroll
        for (int i = 0; i < 64; ++i) {
          int l = i ^ j;
          if (l > i) {
            float av = v[i], bv = v[l];
            float lo = fminf(av, bv), hv = fmaxf(av, bv);
            bool up = ((i & kk) == 0);
            v[i] = up ? lo : hv;
            v[l] = up ? hv : lo;
          }
        }
      }
    }
    float thr = v[64 - 38];   // 38th largest
    float alpha = alpha_p[0];
    int hh = h0 + (t >> 4);
    int ww = w0 + (t & 15);
    size_t base = ((size_t)b * 1024 + d) * 4096 + hh * 64 + ww;
    #pragma unroll 8
    for (int c = 0; c < 64; ++c) {
      float av = attn_s[c * 128 + t];
      size_t gi = base + (size_t)c * (16 * 4096);
      float xv = x[gi];                 // L2-resident re-read
      float res = xv + alpha * ((av >= thr) ? av : 0.0f);
      __builtin_nontemporal_store(res, &out[gi]);  // write-once: don't pollute L2
    }
  }
}

extern "C" void kernel_launch(void* const* d_in, const int* in_sizes, int n_in,
                              void* d_out, int out_size, void* d_ws, size_t ws_size,
                              hipStream_t stream) {
  (void)in_sizes; (void)n_in; (void)d_ws; (void)ws_size; (void)out_size;
  const float* x         = (const float*)d_in[0];
  const float* w_dw      = (const float*)d_in[1];
  const float* bn1_gamma = (const float*)d_in[2];
  const float* bn1_beta  = (const float*)d_in[3];
  const float* bn1_mean  = (const float*)d_in[4];
  const float* bn1_var   = (const float*)d_in[5];
  const float* w_pw1     = (const float*)d_in[6];
  const float* b_pw1     = (const float*)d_in[7];
  const float* bn2_gamma = (const float*)d_in[8];
  const float* bn2_beta  = (const float*)d_in[9];
  const float* bn2_mean  = (const float*)d_in[10];
  const float* bn2_var   = (const float*)d_in[11];
  const float* w_pw2     = (const float*)d_in[12];
  const float* b_pw2     = (const float*)d_in[13];
  const float* alpha_p   = (const float*)d_in[14];
  float* out = (float*)d_out;

  // 4*16 slices * 32 tiles = 2048 blocks, 128 threads each
  lightunit_fused_kernel<<<dim3(2048), dim3(128), 0, stream>>>(
      x, w_dw, bn1_gamma, bn1_beta, bn1_mean, bn1_var,
      w_pw1, b_pw1, bn2_gamma, bn2_beta, bn2_mean, bn2_var,
      w_pw2, b_pw2, alpha_p, out);
}